// GATEncoder_33466385170614
// MI455X (gfx1250) — compile-verified
//
#include <hip/hip_runtime.h>
#include <hip/hip_bf16.h>
#include <math.h>

// ---------------- types for WMMA ----------------
typedef __attribute__((ext_vector_type(2))) float v2f;
typedef __attribute__((ext_vector_type(8))) float v8f;

#define HEADS1 4
#define HID1   64
#define C1     256   // HEADS1*HID1
#define C2     128   // OUT
#define NEG_SLOPE 0.2f

#define KC        128   // K chunk staged in LDS
#define BS_STRIDE 72    // LDS row stride (floats): conflict-free for frag reads

// ---------------- helpers ----------------
__device__ __forceinline__ void atomicMaxF(float* addr, float v) {
    // works with init = -inf; ordered-int-bits trick
    if (v >= 0.0f) atomicMax((int*)addr, __float_as_int(v));
    else           atomicMin((unsigned int*)addr, __float_as_uint(v));
}

__device__ __forceinline__ float lrelu(float v) {
    return v < 0.0f ? NEG_SLOPE * v : v;
}

// ---------------- init kernels ----------------
__global__ void init_f32_k(float* __restrict__ p, size_t n, float val) {
    size_t i = (size_t)blockIdx.x * blockDim.x + threadIdx.x;
    if (i < n) p[i] = val;
}

__global__ void out_bias_k(float* __restrict__ out, const float* __restrict__ b,
                           int n, int C) {
    size_t i = (size_t)blockIdx.x * blockDim.x + threadIdx.x;
    if (i < (size_t)n * C) out[i] = b[i % C];
}

__global__ void bias_relu_k(float* __restrict__ x, const float* __restrict__ b,
                            size_t n, int C) {
    size_t i = (size_t)blockIdx.x * blockDim.x + threadIdx.x;
    if (i < n) {
        float v = x[i] + b[i % C];
        x[i] = v > 0.0f ? v : 0.0f;
    }
}

// ---------------- fp32 WMMA GEMM with LDS-staged B ----------------
// C[M,Ncols] = A[M,K] @ B[K,Ncols].  Block = 4 waves, each wave owns a 16-row
// M tile; all waves share one 64-column B slab staged in LDS in K-chunks of 128.
// Fragment layouts (ISA 7.12.2):
//   A 16x4 f32: lanes 0-15 row=m0+lane K+0/K+1; lanes 16-31 row=m0+lane-16 K+2/K+3
//   B 4x16:     lanes 0-15 col K+0/K+1;          lanes 16-31 col K+2/K+3
//   C/D 16x16:  VGPR r -> (M=r, N=lane) / (M=r+8, N=lane-16)
__global__ void __launch_bounds__(128)
gemm_wmma_f32_k(const float* __restrict__ A,
                const float* __restrict__ B,
                float* __restrict__ C,
                int M, int K, int Ncols) {
    __shared__ float Bs[KC * BS_STRIDE];   // 36 KB

    const int t     = threadIdx.x;         // 0..127
    const int lane  = t & 31;
    const int wave  = t >> 5;
    const int mtile = blockIdx.x * 4 + wave;
    const int m0    = mtile * 16;
    const bool valid = (m0 < M);           // wave-uniform
    const int n0    = blockIdx.y * 64;
    const int l15   = lane & 15;
    const int koff  = (lane >> 4) * 2;     // 0 for lanes 0-15, 2 for lanes 16-31
    const int row   = valid ? (m0 + l15) : 0;

    v8f acc0 = {}, acc1 = {}, acc2 = {}, acc3 = {};

    for (int k0 = 0; k0 < K; k0 += KC) {
        // ---- stage B[k0:k0+KC, n0:n0+64] into LDS (coalesced b128 loads) ----
        __syncthreads();
#pragma unroll
        for (int i = 0; i < (KC * 64 / 4) / 128; ++i) {   // 16 float4 per thread
            int j = t + i * 128;
            int k = j >> 4;                 // row within chunk
            int c = (j & 15) * 4;           // col within 64-wide slab
            const float4 v = *(const float4*)(B + (size_t)(k0 + k) * Ncols + n0 + c);
            *(float4*)(&Bs[k * BS_STRIDE + c]) = v;
        }
        __syncthreads();

        // ---- consume chunk ----
        if (valid) {
            const float* ap0 = A + (size_t)row * K + k0 + koff;
            for (int kk = 0; kk < KC; kk += 4) {
                v2f a; a.x = ap0[kk]; a.y = ap0[kk + 1];

                const float* bp = &Bs[(kk + koff) * BS_STRIDE + l15];
                v2f b0; b0.x = bp[0];  b0.y = bp[BS_STRIDE];
                v2f b1; b1.x = bp[16]; b1.y = bp[BS_STRIDE + 16];
                v2f b2; b2.x = bp[32]; b2.y = bp[BS_STRIDE + 32];
                v2f b3; b3.x = bp[48]; b3.y = bp[BS_STRIDE + 48];

                acc0 = __builtin_amdgcn_wmma_f32_16x16x4_f32(false, a, false, b0, (short)0, acc0, false, false);
                acc1 = __builtin_amdgcn_wmma_f32_16x16x4_f32(false, a, false, b1, (short)0, acc1, false, false);
                acc2 = __builtin_amdgcn_wmma_f32_16x16x4_f32(false, a, false, b2, (short)0, acc2, false, false);
                acc3 = __builtin_amdgcn_wmma_f32_16x16x4_f32(false, a, false, b3, (short)0, acc3, false, false);
            }
        }
    }

    if (valid) {
        const int rbase = m0 + ((lane >> 4) ? 8 : 0);
#pragma unroll
        for (int r = 0; r < 8; ++r) {
            float* cp = C + (size_t)(rbase + r) * Ncols + n0 + l15;
            cp[0]  = acc0[r];
            cp[16] = acc1[r];
            cp[32] = acc2[r];
            cp[48] = acc3[r];
        }
    }
}

// ---------------- attention logits: al[n,h] = <h[n,h,:], a[h,:]> ----------------
__global__ void attn_logits_k(const float* __restrict__ h,
                              const float* __restrict__ a_src,
                              const float* __restrict__ a_dst,
                              float* __restrict__ als, float* __restrict__ ald,
                              int n, int heads, int ch) {
    int i = blockIdx.x * blockDim.x + threadIdx.x;   // over n*heads
    if (i >= n * heads) return;
    int node = i / heads, hd = i % heads;
    const float* hp = h + (size_t)node * heads * ch + (size_t)hd * ch;
    const float* as = a_src + hd * ch;
    const float* ad = a_dst + hd * ch;
    float ss = 0.0f, sd = 0.0f;
    for (int c = 0; c < ch; ++c) {
        float v = hp[c];
        ss += v * as[c];
        sd += v * ad[c];
    }
    als[i] = ss;
    ald[i] = sd;
}

// ---------------- edge pass 1: segment max of leaky-relu logits ----------------
__global__ void edge_max_k(const int* __restrict__ ei, int E, int n,
                           const float* __restrict__ als, const float* __restrict__ ald,
                           float* __restrict__ m, int heads) {
    int e = blockIdx.x * blockDim.x + threadIdx.x;
    int ET = E + n;
    if (e >= ET) return;
    int s, d;
    if (e < E) { s = ei[e]; d = ei[E + e]; } else { s = d = e - E; }
    for (int hd = 0; hd < heads; ++hd) {
        float v = lrelu(als[(size_t)s * heads + hd] + ald[(size_t)d * heads + hd]);
        atomicMaxF(&m[(size_t)d * heads + hd], v);
    }
}

// ---------------- edge pass 2: segment sum of exp(e - m[dst]) ----------------
__global__ void edge_sum_k(const int* __restrict__ ei, int E, int n,
                           const float* __restrict__ als, const float* __restrict__ ald,
                           const float* __restrict__ m, float* __restrict__ den,
                           int heads) {
    int e = blockIdx.x * blockDim.x + threadIdx.x;
    int ET = E + n;
    if (e >= ET) return;
    int s, d;
    if (e < E) { s = ei[e]; d = ei[E + e]; } else { s = d = e - E; }
    for (int hd = 0; hd < heads; ++hd) {
        float v = lrelu(als[(size_t)s * heads + hd] + ald[(size_t)d * heads + hd]);
        float ex = __expf(v - m[(size_t)d * heads + hd]);
        atomicAdd(&den[(size_t)d * heads + hd], ex);
    }
}

// ---------------- edge pass 3: out[dst] += alpha * h[src] ----------------
// blockDim.x == heads*ch (256 for layer1, 128 for layer2); thread = channel.
__global__ void edge_scatter_k(const int* __restrict__ ei, int E, int n,
                               const float* __restrict__ als, const float* __restrict__ ald,
                               const float* __restrict__ m, const float* __restrict__ den,
                               const float* __restrict__ h, float* __restrict__ out,
                               int heads, int ch) {
    const int C  = heads * ch;
    const int t  = threadIdx.x;
    const int hd = t / ch;
    const int ET = E + n;
    for (int e = blockIdx.x; e < ET; e += gridDim.x) {
        int s, d;
        if (e < E) { s = ei[e]; d = ei[E + e]; } else { s = d = e - E; }
        float v = lrelu(als[(size_t)s * heads + hd] + ald[(size_t)d * heads + hd]);
        float alpha = __expf(v - m[(size_t)d * heads + hd]) / den[(size_t)d * heads + hd];
        atomicAdd(&out[(size_t)d * C + t], h[(size_t)s * C + t] * alpha);
    }
}

// ---------------- launcher ----------------
extern "C" void kernel_launch(void* const* d_in, const int* in_sizes, int n_in,
                              void* d_out, int out_size, void* d_ws, size_t ws_size,
                              hipStream_t stream) {
    const float* x      = (const float*)d_in[0];   // [N,128]
    const int*   eidx   = (const int*)d_in[1];     // [2,E]
    const float* W1     = (const float*)d_in[2];   // [128,256]
    const float* a1_src = (const float*)d_in[3];   // [4,64]
    const float* a1_dst = (const float*)d_in[4];
    const float* b1     = (const float*)d_in[5];   // [256]
    const float* W2     = (const float*)d_in[6];   // [256,128]
    const float* a2_src = (const float*)d_in[7];   // [1,128]
    const float* a2_dst = (const float*)d_in[8];
    const float* b2     = (const float*)d_in[9];   // [128]
    float*       out    = (float*)d_out;           // [N,128]

    const int N = in_sizes[0] / 128;
    const int E = in_sizes[1] / 2;
    const int ET = E + N;

    // workspace layout (floats)
    float* ws   = (float*)d_ws;
    float* h1   = ws;                               // N*256 (reused as h2: N*128)
    float* out1 = h1   + (size_t)N * C1;            // N*256 (becomes x2 after relu)
    float* als1 = out1 + (size_t)N * C1;            // N*4
    float* ald1 = als1 + (size_t)N * HEADS1;
    float* m1   = ald1 + (size_t)N * HEADS1;
    float* den1 = m1   + (size_t)N * HEADS1;
    float* als2 = den1 + (size_t)N * HEADS1;        // N
    float* ald2 = als2 + (size_t)N;
    float* m2   = ald2 + (size_t)N;
    float* den2 = m2   + (size_t)N;
    float* h2   = h1;                               // reuse

    const int mtiles = (N + 15) / 16;

    // ===== Layer 1 =====
    {   // h1 = x @ W1  (M=N, K=128, Ncols=256)
        dim3 grid((mtiles + 3) / 4, C1 / 64);
        gemm_wmma_f32_k<<<grid, 128, 0, stream>>>(x, W1, h1, N, 128, C1);
    }
    {   // attention logits
        int tot = N * HEADS1;
        attn_logits_k<<<(tot + 255) / 256, 256, 0, stream>>>(h1, a1_src, a1_dst,
                                                             als1, ald1, N, HEADS1, HID1);
    }
    {   // init stats + output accumulator
        size_t nm = (size_t)N * HEADS1;
        init_f32_k<<<(int)((nm + 255) / 256), 256, 0, stream>>>(m1, nm, -INFINITY);
        init_f32_k<<<(int)((nm + 255) / 256), 256, 0, stream>>>(den1, nm, 0.0f);
        size_t no = (size_t)N * C1;
        init_f32_k<<<(int)((no + 255) / 256), 256, 0, stream>>>(out1, no, 0.0f);
    }
    edge_max_k<<<(ET + 255) / 256, 256, 0, stream>>>(eidx, E, N, als1, ald1, m1, HEADS1);
    edge_sum_k<<<(ET + 255) / 256, 256, 0, stream>>>(eidx, E, N, als1, ald1, m1, den1, HEADS1);
    edge_scatter_k<<<ET, C1, 0, stream>>>(eidx, E, N, als1, ald1, m1, den1, h1, out1,
                                          HEADS1, HID1);
    {   // x2 = relu(out1 + b1) in place
        size_t no = (size_t)N * C1;
        bias_relu_k<<<(int)((no + 255) / 256), 256, 0, stream>>>(out1, b1, no, C1);
    }

    // ===== Layer 2 =====
    {   // h2 = x2 @ W2  (M=N, K=256, Ncols=128)
        dim3 grid((mtiles + 3) / 4, C2 / 64);
        gemm_wmma_f32_k<<<grid, 128, 0, stream>>>(out1, W2, h2, N, C1, C2);
    }
    attn_logits_k<<<(N + 255) / 256, 256, 0, stream>>>(h2, a2_src, a2_dst,
                                                       als2, ald2, N, 1, C2);
    {
        init_f32_k<<<(N + 255) / 256, 256, 0, stream>>>(m2, (size_t)N, -INFINITY);
        init_f32_k<<<(N + 255) / 256, 256, 0, stream>>>(den2, (size_t)N, 0.0f);
        size_t no = (size_t)N * C2;
        out_bias_k<<<(int)((no + 255) / 256), 256, 0, stream>>>(out, b2, N, C2);
    }
    edge_max_k<<<(ET + 255) / 256, 256, 0, stream>>>(eidx, E, N, als2, ald2, m2, 1);
    edge_sum_k<<<(ET + 255) / 256, 256, 0, stream>>>(eidx, E, N, als2, ald2, m2, den2, 1);
    edge_scatter_k<<<ET, C2, 0, stream>>>(eidx, E, N, als2, ald2, m2, den2, h2, out,
                                          1, C2);
}